// GIN_23845658427623
// MI455X (gfx1250) — compile-verified
//
#include <hip/hip_runtime.h>
#include <hip/hip_bf16.h>

typedef __attribute__((ext_vector_type(2))) float v2f;
typedef __attribute__((ext_vector_type(8))) float v8f;

#define GIN_D 128
#define LDSP  132   // padded LDS row stride (floats): 132*4B = 528B, 16B-aligned, bank-conflict-free

// ---------------------------------------------------------------- zero (float4-wide)
__global__ void gin_zero4_kernel(float4* __restrict__ p, int n4) {
    int i = blockIdx.x * blockDim.x + threadIdx.x;
    if (i < n4) p[i] = make_float4(0.f, 0.f, 0.f, 0.f);
}

// ---------------------------------------------------------------- edge scatter-add
// One wave per edge: lane l owns columns 4l..4l+3. b128 gather of h[src] +
// four coalesced 4B float atomics into agg[dst] (L2-resident: h+agg = 102MB < 192MB L2).
__global__ void gin_scatter_kernel(const float* __restrict__ h,
                                   const int* __restrict__ src,
                                   const int* __restrict__ dst,
                                   float* __restrict__ agg, int n_edges) {
    int lane   = threadIdx.x & 31;
    int egroup = threadIdx.x >> 5;            // 0..7 (256 threads = 8 waves)
    int c0     = lane * 4;
    for (int e = blockIdx.x * 8 + egroup; e < n_edges; e += gridDim.x * 8) {
        int s = src[e];
        int t = dst[e];
        float4 v = *(const float4*)(h + s * GIN_D + c0);
        float* a = agg + t * GIN_D + c0;
        atomicAdd(a + 0, v.x);
        atomicAdd(a + 1, v.y);
        atomicAdd(a + 2, v.z);
        atomicAdd(a + 3, v.w);
    }
}

// ---------------------------------------------------------------- fused (h+agg)@W + b, relu
// Block = 256 threads = 8 waves. Block owns a 16-row stripe; wave w owns the
// 16-col tile n0 = 16*w. W staged TRANSPOSED in LDS so both A and B operands of
// V_WMMA_F32_16X16X4_F32 are contiguous 8B pairs -> ds_load_2addr_b64 feeds,
// zero global traffic in the K loop.
__global__ void __launch_bounds__(256)
gin_gemm_relu_kernel(const float* __restrict__ h,
                     const float* __restrict__ agg,
                     const float* __restrict__ W,     // [128,128] row-major
                     const float* __restrict__ bias,  // [128]
                     float* __restrict__ out, int n_nodes) {
    __shared__ float Xs[16 * LDSP];            // x = h+agg tile, row-major
    __shared__ float Wt[GIN_D * LDSP];         // W transposed: Wt[n*LDSP + k] = W[k][n]

    const int tid  = threadIdx.x;
    const int row0 = blockIdx.x * 16;

    // Stage x-tile (16x128) with float4 loads/stores.
    for (int idx = tid; idx < 16 * (GIN_D / 4); idx += 256) {
        int r  = idx >> 5;                     // 0..15
        int c4 = (idx & 31) * 4;               // 0,4,..,124
        int row = row0 + r;
        float4 v = make_float4(0.f, 0.f, 0.f, 0.f);
        if (row < n_nodes) {
            float4 a = *(const float4*)(h   + row * GIN_D + c4);
            float4 g = *(const float4*)(agg + row * GIN_D + c4);
            v = make_float4(a.x + g.x, a.y + g.y, a.z + g.z, a.w + g.w);
        }
        *(float4*)(&Xs[r * LDSP + c4]) = v;    // r*LDSP*4B = r*528B, 16B aligned
    }
    // Stage W transposed (coalesced float4 global reads, scattered 4B LDS writes).
    for (int idx = tid; idx < GIN_D * (GIN_D / 4); idx += 256) {
        int k  = idx >> 5;                     // 0..127
        int n4 = (idx & 31) * 4;               // 0,4,..,124
        float4 w = *(const float4*)(W + k * GIN_D + n4);
        Wt[(n4 + 0) * LDSP + k] = w.x;
        Wt[(n4 + 1) * LDSP + k] = w.y;
        Wt[(n4 + 2) * LDSP + k] = w.z;
        Wt[(n4 + 3) * LDSP + k] = w.w;
    }
    __syncthreads();

    const int wave = tid >> 5;                 // 0..7 -> column tile
    const int lane = tid & 31;
    const int half = lane >> 4;                // 0: lanes 0-15, 1: lanes 16-31
    const int m    = lane & 15;                // A-matrix row
    const int n    = lane & 15;                // B/C/D column within tile
    const int col  = wave * 16 + n;

    const float* xrow = &Xs[m * LDSP];
    const float* wcol = &Wt[col * LDSP];

    v8f acc = {};
#pragma unroll
    for (int kk = 0; kk < GIN_D; kk += 4) {
        const int kb = kk + 2 * half;          // per ISA 16x4 f32 A/B K-slot layout
        v2f a, b;
        a.x = xrow[kb + 0];
        a.y = xrow[kb + 1];
        b.x = wcol[kb + 0];
        b.y = wcol[kb + 1];
        acc = __builtin_amdgcn_wmma_f32_16x16x4_f32(
            false, a, false, b, (short)0, acc, false, false);
    }

    const float bv = bias[col];
#pragma unroll
    for (int v = 0; v < 8; ++v) {
        int row = row0 + v + 8 * half;         // C/D layout: VGPR v -> M = v (+8 hi half)
        if (row < n_nodes) {
            float val = acc[v] + bv;
            out[row * GIN_D + col] = val > 0.0f ? val : 0.0f;
        }
    }
}

// ---------------------------------------------------------------- per-graph sum pool
// graph_ids is sorted: accumulate locally per 16-node chunk, flush one atomic
// per (graph, column) transition.
__global__ void gin_pool_kernel(const float* __restrict__ h,
                                const int* __restrict__ gid,
                                float* __restrict__ sums,   // [G,128]
                                float* __restrict__ counts, // [G]
                                int n_nodes) {
    int tid  = threadIdx.x;                    // 0..127 (column)
    int base = blockIdx.x * 16;
    if (base >= n_nodes) return;
    int end = base + 16; if (end > n_nodes) end = n_nodes;

    int cur_g = gid[base];
    float acc = 0.0f;
    int cnt = 0;
    for (int i = base; i < end; ++i) {
        int g = gid[i];
        if (g != cur_g) {
            atomicAdd(&sums[cur_g * GIN_D + tid], acc);
            if (tid == 0) atomicAdd(&counts[cur_g], (float)cnt);
            acc = 0.0f; cnt = 0; cur_g = g;
        }
        acc += h[i * GIN_D + tid];
        cnt++;
    }
    atomicAdd(&sums[cur_g * GIN_D + tid], acc);
    if (tid == 0) atomicAdd(&counts[cur_g], (float)cnt);
}

// ---------------------------------------------------------------- mean + tiny FC
__global__ void gin_fc_kernel(const float* __restrict__ sums,
                              const float* __restrict__ counts,
                              const float* __restrict__ Wfc,  // [128, NC]
                              const float* __restrict__ bfc,  // [NC]
                              float* __restrict__ out, int total, int nc) {
    int t = blockIdx.x * blockDim.x + threadIdx.x;
    if (t >= total) return;
    int g = t / nc;
    int c = t % nc;
    float inv = 1.0f / fmaxf(counts[g], 1.0f);
    float s = 0.0f;
    for (int k = 0; k < GIN_D; ++k)
        s += (sums[g * GIN_D + k] * inv) * Wfc[k * nc + c];
    out[t] = s + bfc[c];
}

// ---------------------------------------------------------------- driver
extern "C" void kernel_launch(void* const* d_in, const int* in_sizes, int n_in,
                              void* d_out, int out_size, void* d_ws, size_t ws_size,
                              hipStream_t stream) {
    const float* in_feat = (const float*)d_in[0];
    const int*   src     = (const int*)d_in[1];
    const int*   dst     = (const int*)d_in[2];
    const int*   gid     = (const int*)d_in[3];
    const float* W1      = (const float*)d_in[4];
    const float* b1      = (const float*)d_in[5];
    const float* W2      = (const float*)d_in[6];
    const float* b2      = (const float*)d_in[7];
    const float* W3      = (const float*)d_in[8];
    const float* b3      = (const float*)d_in[9];
    const float* Wfc     = (const float*)d_in[10];
    const float* bfc     = (const float*)d_in[11];

    const int N  = in_sizes[0] / GIN_D;        // 100000
    const int E  = in_sizes[1];                // 1600000
    const int NC = in_sizes[11];               // 2
    const int NG = out_size / NC;              // 64

    float* agg    = (float*)d_ws;
    float* hA     = agg + (size_t)N * GIN_D;
    float* hB     = hA  + (size_t)N * GIN_D;
    float* sums   = hB  + (size_t)N * GIN_D;   // [NG,128] then counts[NG]
    float* counts = sums + (size_t)NG * GIN_D;

    const int nd    = N * GIN_D;
    const dim3 zgrid((nd / 4 + 255) / 256);
    const int  ggrid = (N + 15) / 16;

    float* out = (float*)d_out;

    // ---- layer 1
    gin_zero4_kernel<<<zgrid, 256, 0, stream>>>((float4*)agg, nd / 4);
    gin_scatter_kernel<<<8192, 256, 0, stream>>>(in_feat, src, dst, agg, E);
    gin_gemm_relu_kernel<<<ggrid, 256, 0, stream>>>(in_feat, agg, W1, b1, hA, N);
    // ---- layer 2
    gin_zero4_kernel<<<zgrid, 256, 0, stream>>>((float4*)agg, nd / 4);
    gin_scatter_kernel<<<8192, 256, 0, stream>>>(hA, src, dst, agg, E);
    gin_gemm_relu_kernel<<<ggrid, 256, 0, stream>>>(hA, agg, W2, b2, hB, N);
    // ---- layer 3
    gin_zero4_kernel<<<zgrid, 256, 0, stream>>>((float4*)agg, nd / 4);
    gin_scatter_kernel<<<8192, 256, 0, stream>>>(hB, src, dst, agg, E);
    gin_gemm_relu_kernel<<<ggrid, 256, 0, stream>>>(hB, agg, W3, b3, hA, N);

    // ---- pool + FC (zero sums+counts together: counts is contiguous after sums)
    gin_zero4_kernel<<<1, 256, 0, stream>>>((float4*)sums, (NG * GIN_D + NG) / 4);
    gin_pool_kernel<<<ggrid, GIN_D, 0, stream>>>(hA, gid, sums, counts, N);
    gin_fc_kernel<<<(out_size + 127) / 128, 128, 0, stream>>>(sums, counts, Wfc, bfc, out, out_size, NC);
}